// YibeiVessels_74156905333425
// MI455X (gfx1250) — compile-verified
//
#include <hip/hip_runtime.h>
#include <stdint.h>

// ---------------------------------------------------------------------------
// YibeiVessels: per-voxel LUT gather + elementwise scale/one-hot over 256^3.
// Bandwidth-bound (≈384 MB traffic, ≈16.5 us at 23.3 TB/s). No matrix math ->
// no WMMA; optimize for b128 ld/st, NT cache hints, LDS LUT (ds_load_b64),
// and gfx1250 async global->LDS for the LUT fill.
// ---------------------------------------------------------------------------

#define DVOX 256
#define NVOX (DVOX * DVOX * DVOX)   // 16,777,216
#define NIDS 257
#define TPB 256
#define VPT 4                        // voxels per thread (b128 ld/st)

typedef int   v4i __attribute__((ext_vector_type(4)));
typedef float v4f __attribute__((ext_vector_type(4)));

struct __align__(8) LutEntry {
    float    inten;
    uint32_t cls;
};

__global__ __launch_bounds__(TPB) void
YibeiVessels_74156905333425_kernel(const int*   __restrict__ labels,
                                   const float* __restrict__ intensities,
                                   const int*   __restrict__ class_bits,
                                   const float* __restrict__ parenchyma,
                                   float*       __restrict__ out)
{
    // 257-entry LUT, 8 bytes/id -> one ds_load_b64 per voxel gather.
    __shared__ LutEntry lut[NIDS + 7];

#if __has_builtin(__builtin_amdgcn_global_load_async_to_lds_b32) && \
    __has_builtin(__builtin_amdgcn_s_wait_asynccnt)
    // gfx1250 async global->LDS path for the cooperative LUT fill.
    {
        const int i = threadIdx.x;
        if (i < NIDS) {
            __builtin_amdgcn_global_load_async_to_lds_b32(
                (int*)(intensities + i), (int*)&lut[i].inten, 0, 0);
            __builtin_amdgcn_global_load_async_to_lds_b32(
                (int*)(class_bits + i), (int*)&lut[i].cls, 0, 0);
        }
        // blockDim == 256 < 257: thread 0 also covers id 256.
        if (i == 0) {
            __builtin_amdgcn_global_load_async_to_lds_b32(
                (int*)(intensities + 256), (int*)&lut[256].inten, 0, 0);
            __builtin_amdgcn_global_load_async_to_lds_b32(
                (int*)(class_bits + 256), (int*)&lut[256].cls, 0, 0);
        }
        __builtin_amdgcn_s_wait_asynccnt(0);
    }
#else
    for (int i = threadIdx.x; i < NIDS; i += TPB) {
        LutEntry e;
        e.inten = intensities[i];
        e.cls   = (uint32_t)class_bits[i];
        lut[i]  = e;
    }
#endif
    __syncthreads();

    const size_t base = ((size_t)blockIdx.x * TPB + (size_t)threadIdx.x) * VPT;

    // Streaming reads: read-once, keep out of near caches (th:NT).
    v4i lab = __builtin_nontemporal_load((const v4i*)(labels + base));
    v4f par = __builtin_nontemporal_load((const v4f*)(parenchyma + base));

    v4f vol, c0, c1, c2;
#pragma unroll
    for (int k = 0; k < VPT; ++k) {
        const int  l      = lab[k];
        const LutEntry e  = lut[l];           // single ds_load_b64
        const bool vessel = (l > 0);
        const float s     = vessel ? e.inten : 1.0f;
        vol[k] = par[k] * s;
        c0[k]  = vessel ? 0.0f : 1.0f;
        c1[k]  = (vessel && (e.cls == 0u)) ? 1.0f : 0.0f;
        c2[k]  = (vessel && (e.cls == 1u)) ? 1.0f : 0.0f;
    }

    // Output layout: [out_vol (N), ch0 (N), ch1 (N), ch2 (N)].
    // 256 MB write-once stream > 192 MB L2 -> non-temporal b128 stores.
    float* o = out + base;
    __builtin_nontemporal_store(vol, (v4f*)(o));
    __builtin_nontemporal_store(c0,  (v4f*)(o + (size_t)NVOX));
    __builtin_nontemporal_store(c1,  (v4f*)(o + 2 * (size_t)NVOX));
    __builtin_nontemporal_store(c2,  (v4f*)(o + 3 * (size_t)NVOX));
}

extern "C" void kernel_launch(void* const* d_in, const int* in_sizes, int n_in,
                              void* d_out, int out_size, void* d_ws, size_t ws_size,
                              hipStream_t stream)
{
    const int*   labels      = (const int*)  d_in[0];
    const float* intensities = (const float*)d_in[1];
    const int*   class_bits  = (const int*)  d_in[2];
    const float* parenchyma  = (const float*)d_in[3];
    float*       out         = (float*)      d_out;

    const int blocks = NVOX / (TPB * VPT);   // 16384, exact
    YibeiVessels_74156905333425_kernel<<<blocks, TPB, 0, stream>>>(
        labels, intensities, class_bits, parenchyma, out);
}